// InstanceContrastiveLoss_13855564496989
// MI455X (gfx1250) — compile-verified
//
#include <hip/hip_runtime.h>
#include <hip/hip_bf16.h>
#include <climits>

// ---------------------------------------------------------------------------
// InstanceContrastiveLoss for MI455X (gfx1250, wave32).
//   - mask scans: LDS-pre-reduced atomics + global prefetch (memory bound, ~8MB)
//   - triplet distances: V_WMMA_F32_16X16X4_F32 Gram diagonals (full f32,
//     matches reference numerics; A and A^T share one register layout)
//   - negative sampling: threefry2x32 replay of jax.random.key(1)
// ---------------------------------------------------------------------------

#define KIDS    64
#define NCANDS  16
#define MARGINF 1.0f
#define EPSF    1e-6f

typedef float v2f __attribute__((ext_vector_type(2)));
typedef float v8f __attribute__((ext_vector_type(8)));

#if __has_builtin(__builtin_amdgcn_wmma_f32_16x16x4_f32)
#define HAVE_WMMA_F32X4 1
#else
#define HAVE_WMMA_F32X4 0
#endif

// Classic jax threefry2x32, key = PRNGKey(1) = (0, 1).
__device__ __forceinline__ void icl_threefry2x32(unsigned x0, unsigned x1,
                                                 unsigned& o0, unsigned& o1) {
  const unsigned k0 = 0u, k1 = 1u;
  const unsigned k2 = k0 ^ k1 ^ 0x1BD11BDAu;
  x0 += k0; x1 += k1;
#define TF_R(r) { x0 += x1; x1 = (x1 << (r)) | (x1 >> (32 - (r))); x1 ^= x0; }
  TF_R(13) TF_R(15) TF_R(26) TF_R(6)   x0 += k1; x1 += k2 + 1u;
  TF_R(17) TF_R(29) TF_R(16) TF_R(24)  x0 += k2; x1 += k0 + 2u;
  TF_R(13) TF_R(15) TF_R(26) TF_R(6)   x0 += k0; x1 += k1 + 3u;
  TF_R(17) TF_R(29) TF_R(16) TF_R(24)  x0 += k1; x1 += k2 + 4u;
  TF_R(13) TF_R(15) TF_R(26) TF_R(6)   x0 += k2; x1 += k0 + 5u;
#undef TF_R
  o0 = x0; o1 = x1;
}

__global__ void icl_init_kernel(int* __restrict__ firstg, int* __restrict__ secondg,
                                int* __restrict__ countsg) {
  const int t = threadIdx.x;
  if (t < KIDS) { firstg[t] = INT_MAX; secondg[t] = INT_MAX; countsg[t] = 0; }
}

// Pass 1: first occurrence (segment_min over pixel index) + counts.
__global__ void icl_scan1_kernel(const int* __restrict__ mask, int* __restrict__ firstg,
                                 int* __restrict__ countsg, int HW) {
  __shared__ int lf[KIDS];
  __shared__ int lc[KIDS];
  const int t = threadIdx.x;
  if (t < KIDS) { lf[t] = INT_MAX; lc[t] = 0; }
  __syncthreads();
  const int stride = (int)(gridDim.x * blockDim.x);
  for (int i = (int)(blockIdx.x * blockDim.x) + t; i < HW; i += stride) {
    if (i + stride < HW) __builtin_prefetch(&mask[i + stride], 0, 0);  // global_prefetch_b8
    const int id = mask[i];
    atomicMin(&lf[id], i);
    atomicAdd(&lc[id], 1);
  }
  __syncthreads();
  if (t < KIDS) {
    if (lf[t] != INT_MAX) atomicMin(&firstg[t], lf[t]);
    if (lc[t] != 0)       atomicAdd(&countsg[t], lc[t]);
  }
}

// Pass 2: second occurrence = min over pixels whose index != first[id].
__global__ void icl_scan2_kernel(const int* __restrict__ mask, const int* __restrict__ firstg,
                                 int* __restrict__ secondg, int HW) {
  __shared__ int ls[KIDS];
  __shared__ int lfirst[KIDS];
  const int t = threadIdx.x;
  if (t < KIDS) { ls[t] = INT_MAX; lfirst[t] = firstg[t]; }
  __syncthreads();
  const int stride = (int)(gridDim.x * blockDim.x);
  for (int i = (int)(blockIdx.x * blockDim.x) + t; i < HW; i += stride) {
    if (i + stride < HW) __builtin_prefetch(&mask[i + stride], 0, 0);
    const int id = mask[i];
    if (i != lfirst[id]) atomicMin(&ls[id], i);
  }
  __syncthreads();
  if (t < KIDS) {
    if (ls[t] != INT_MAX) atomicMin(&secondg[t], ls[t]);
  }
}

// Final kernel: one block, 4 waves; wave w owns instances [16w, 16w+16).
// Per wave: accumulate C = X * X^T with V_WMMA_F32_16X16X4_F32 over the
// 76-channel K dimension; diag(C) = squared triplet distances.
__global__ void icl_loss_kernel(const float* __restrict__ sem,
                                const int* __restrict__ mask,
                                const int* __restrict__ firstg,
                                const int* __restrict__ secondg,
                                const int* __restrict__ countsg,
                                float* __restrict__ out,
                                int HW, int CH) {
  __shared__ int   sF[KIDS], sS[KIDS], sN[KIDS];
  __shared__ float sValid[KIDS];
  __shared__ float tileA[4][16][17];   // per-wave C staging (anchor-positive)
  __shared__ float tileB[4][16][17];   // per-wave C staging (anchor-negative)
  __shared__ float dap2[KIDS], dan2[KIDS];

  const int tid = (int)threadIdx.x;

  // ---- negative sampling + per-instance metadata (threads 0..63) ----------
  if (tid < KIDS) {
    const int id = tid;
    int neg = -1, cand0 = 0;
    const int half = (KIDS * NCANDS) / 2;   // 512
    for (int j = 0; j < NCANDS; ++j) {
      const int idx = id * NCANDS + j;      // flat index into (K, NCAND)
      unsigned o0, o1, bits;
      if (idx < half) { icl_threefry2x32((unsigned)idx, (unsigned)(idx + half), o0, o1); bits = o0; }
      else            { icl_threefry2x32((unsigned)(idx - half), (unsigned)idx, o0, o1); bits = o1; }
      const int cand = (int)(bits % (unsigned)HW);   // HW = 2^20 -> low bits
      if (j == 0) cand0 = cand;
      if (neg < 0 && mask[cand] != id) neg = cand;   // argmax = first true
    }
    if (neg < 0) neg = cand0;                        // all-false -> index 0
    int f = firstg[id];
    int s = secondg[id];
    f = (f < 0) ? 0 : ((f > HW - 1) ? (HW - 1) : f); // jnp.clip
    s = (s < 0) ? 0 : ((s > HW - 1) ? (HW - 1) : s);
    sF[id] = f; sS[id] = s; sN[id] = neg;
    sValid[id] = (countsg[id] >= 2 && id != 0) ? 1.0f : 0.0f;
  }
  __syncthreads();

  // ---- WMMA Gram-diagonal distance accumulation ---------------------------
  const int lane  = tid & 31;
  const int wave  = tid >> 5;
  const int m     = lane & 15;        // matrix row this lane carries
  const int khalf = lane >> 4;        // K-pair select (A layout: v0=K0/K2, v1=K1/K3)
  const int inst  = wave * 16 + m;
  const int fp = sF[inst], sp = sS[inst], np = sN[inst];

#if HAVE_WMMA_F32X4
  v8f cap = {0.f,0.f,0.f,0.f,0.f,0.f,0.f,0.f};
  v8f can = {0.f,0.f,0.f,0.f,0.f,0.f,0.f,0.f};
#else
  float pap = 0.f, pan = 0.f;
#endif

  const int nstep = CH >> 2;          // CH = 76 -> 19 K-steps of 4
  for (int t = 0; t < nstep; ++t) {
    const int c0 = t * 4 + khalf * 2;
    const float* base0 = sem + (size_t)c0 * (size_t)HW;
    const float* base1 = base0 + (size_t)HW;
    const float a0 = base0[fp], a1 = base1[fp];
    const float p0 = base0[sp], p1 = base1[sp];
    const float n0 = base0[np], n1 = base1[np];
    v2f xap = { a0 - p0 + EPSF, a1 - p1 + EPSF };
    v2f xan = { a0 - n0 + EPSF, a1 - n1 + EPSF };
#if HAVE_WMMA_F32X4
    // A (16x4) and B=A^T (4x16) occupy identical per-lane registers -> pass x twice.
    cap = __builtin_amdgcn_wmma_f32_16x16x4_f32(false, xap, false, xap, (short)0, cap, false, false);
    can = __builtin_amdgcn_wmma_f32_16x16x4_f32(false, xan, false, xan, (short)0, can, false, false);
#else
    pap += xap.x * xap.x + xap.y * xap.y;
    pan += xan.x * xan.x + xan.y * xan.y;
#endif
  }

#if HAVE_WMMA_F32X4
  // C/D layout: VGPR j holds row (j + 8*khalf), column = lane&15.
#pragma unroll
  for (int j = 0; j < 8; ++j) {
    tileA[wave][j + khalf * 8][m] = cap[j];
    tileB[wave][j + khalf * 8][m] = can[j];
  }
#else
  tileA[wave][m][khalf] = pap;
  tileB[wave][m][khalf] = pan;
#endif
  __syncthreads();

  // ---- diagonal extraction + K-tail (CH % 4, none for CH=76) --------------
  if (lane < 16) {
    const int i2 = wave * 16 + lane;
#if HAVE_WMMA_F32X4
    float ap2 = tileA[wave][lane][lane];
    float an2 = tileB[wave][lane][lane];
#else
    float ap2 = tileA[wave][lane][0] + tileA[wave][lane][1];
    float an2 = tileB[wave][lane][0] + tileB[wave][lane][1];
#endif
    const int f2 = sF[i2], s2 = sS[i2], n2 = sN[i2];
    for (int c = nstep * 4; c < CH; ++c) {
      const float* b = sem + (size_t)c * (size_t)HW;
      const float da = b[f2] - b[s2] + EPSF; ap2 += da * da;
      const float dn = b[f2] - b[n2] + EPSF; an2 += dn * dn;
    }
    dap2[i2] = ap2;
    dan2[i2] = an2;
  }
  __syncthreads();

  // ---- deterministic serial reduction (64 values) -------------------------
  if (tid == 0) {
    float total = 0.f, cnt = 0.f;
    for (int i = 0; i < KIDS; ++i) {
      const float per = sqrtf(dap2[i]) - sqrtf(dan2[i]) + MARGINF;
      const float hinge = per > 0.f ? per : 0.f;
      total += sValid[i] * hinge;
      cnt   += sValid[i];
    }
    out[0] = (cnt > 0.f) ? (total / cnt) : 0.f;
  }
}

extern "C" void kernel_launch(void* const* d_in, const int* in_sizes, int n_in,
                              void* d_out, int out_size, void* d_ws, size_t ws_size,
                              hipStream_t stream) {
  const float* sem  = (const float*)d_in[0];   // (4,19,1024,1024) f32
  const int*   mask = (const int*)d_in[1];     // (1024,1024) i32 in [0,64)
  const int HW = in_sizes[1];
  const int CH = in_sizes[0] / HW;             // B*C = 76

  int* firstg  = (int*)d_ws;
  int* secondg = firstg + KIDS;
  int* countsg = firstg + 2 * KIDS;

  icl_init_kernel<<<1, 64, 0, stream>>>(firstg, secondg, countsg);
  icl_scan1_kernel<<<256, 256, 0, stream>>>(mask, firstg, countsg, HW);
  icl_scan2_kernel<<<256, 256, 0, stream>>>(mask, firstg, secondg, HW);
  icl_loss_kernel<<<1, 128, 0, stream>>>(sem, mask, firstg, secondg, countsg,
                                         (float*)d_out, HW, CH);
}